// GATWithMemPooling_47614007444091
// MI455X (gfx1250) — compile-verified
//
#include <hip/hip_runtime.h>
#include <math.h>

#define N_NODES   32768
#define NUM_EDGES 524288
#define E_TOT     (NUM_EDGES + N_NODES)   // 557056 incl. self loops
#define NUM_GRAPHS 32
#define PER_G     (N_NODES / NUM_GRAPHS)  // 1024
#define C         128
#define N_LAYERS  3
#define D_IN      144
#define C_MID     80
#define C_OUT     64
#define H1        5
#define K1N       10
#define NC1       (H1 * K1N)              // 50 clusters
#define NC1P      64                      // padded to multiple of 16
#define KL_EPS    1e-15f

typedef __attribute__((ext_vector_type(2))) float v2f;
typedef __attribute__((ext_vector_type(4))) float v4f;
typedef __attribute__((ext_vector_type(8))) float v8f;

// ---------------------------------------------------------------------------
// Register-blocked fp32 WMMA GEMM: one wave computes a 64x32 C macro-tile as
// a 4x2 grid of 16x16 V_WMMA_F32_16X16X4_F32 tiles.  8 v8f accumulators
// (64 VGPRs) + 6 frags + addressing ~= 100 VGPRs -> no scratch spills (the
// 4x4 variant spilled accumulators around every WMMA).  Per K-step: 12
// lane-loads feed 8 WMMAs.  M mult of 64, N mult of 32, K mult of 4.
// ---------------------------------------------------------------------------
__global__ void __launch_bounds__(256, 1)
wmma_gemm_f32_t64x32(const float* __restrict__ A,
                     const float* __restrict__ B,
                     const float* __restrict__ bias,
                     float* __restrict__ Cm,
                     int M, int K, int N)
{
    int wave = (blockIdx.x * blockDim.x + threadIdx.x) >> 5;
    int lane = threadIdx.x & 31;
    int ntiles = N >> 5;                       // 32-wide macro-tiles
    int tiles  = (M >> 6) * ntiles;
    if (wave >= tiles) return;
    int tm = (wave / ntiles) << 6;
    int tn = (wave % ntiles) << 5;
    int half = lane >> 4;
    int r    = lane & 15;

    v8f acc[8];
#pragma unroll
    for (int i = 0; i < 8; ++i) acc[i] = (v8f){0.f,0.f,0.f,0.f,0.f,0.f,0.f,0.f};

    const float* arow0 = A + (size_t)(tm + r) * K;
    const float* bcol0 = B + (tn + r);
    for (int k0 = 0; k0 < K; k0 += 4) {
        v2f a[4], b[2];
#pragma unroll
        for (int mi = 0; mi < 4; ++mi) {
            const float* ar = arow0 + (size_t)(16 * mi) * K;
            a[mi].x = ar[k0 + 2 * half];
            a[mi].y = ar[k0 + 2 * half + 1];
        }
#pragma unroll
        for (int ni = 0; ni < 2; ++ni) {
            const float* bc = bcol0 + 16 * ni;
            b[ni].x = bc[(size_t)(k0 + 2 * half) * N];
            b[ni].y = bc[(size_t)(k0 + 2 * half + 1) * N];
        }
#pragma unroll
        for (int mi = 0; mi < 4; ++mi)
#pragma unroll
            for (int ni = 0; ni < 2; ++ni)
                acc[mi * 2 + ni] = __builtin_amdgcn_wmma_f32_16x16x4_f32(
                    false, a[mi], false, b[ni], (short)0, acc[mi * 2 + ni],
                    false, false);
    }
#pragma unroll
    for (int mi = 0; mi < 4; ++mi) {
#pragma unroll
        for (int ni = 0; ni < 2; ++ni) {
            int col = tn + 16 * ni + r;
            float bv = bias ? bias[col] : 0.0f;
#pragma unroll
            for (int i = 0; i < 8; ++i) {
                int row = tm + 16 * mi + i + half * 8;
                Cm[(size_t)row * N + col] = acc[mi * 2 + ni][i] + bv;
            }
        }
    }
}

// ---------------------------------------------------------------------------
// Generic fp32 WMMA GEMM (16x16 tile per wave) for shapes that are only
// multiples of 16 (used for the small 320x128x80 mem-pool GEMM).
// ---------------------------------------------------------------------------
__global__ void wmma_gemm_f32(const float* __restrict__ A,
                              const float* __restrict__ B,
                              const float* __restrict__ bias,
                              float* __restrict__ Cm,
                              int M, int K, int N, int act)
{
    int wave = (blockIdx.x * blockDim.x + threadIdx.x) >> 5;
    int lane = threadIdx.x & 31;
    int ntiles = N >> 4;
    int tiles  = (M >> 4) * ntiles;
    if (wave >= tiles) return;
    int tm = (wave / ntiles) << 4;
    int tn = (wave % ntiles) << 4;
    int half = lane >> 4;
    int r    = lane & 15;

    v8f acc = {0.f, 0.f, 0.f, 0.f, 0.f, 0.f, 0.f, 0.f};
    const float* arow = A + (size_t)(tm + r) * K;
    const float* bcol = B + (tn + r);
    for (int k0 = 0; k0 < K; k0 += 4) {
        v2f a, b;
        a.x = arow[k0 + 2 * half];
        a.y = arow[k0 + 2 * half + 1];
        b.x = bcol[(size_t)(k0 + 2 * half) * N];
        b.y = bcol[(size_t)(k0 + 2 * half + 1) * N];
        acc = __builtin_amdgcn_wmma_f32_16x16x4_f32(false, a, false, b,
                                                    (short)0, acc, false, false);
    }
    int col = tn + r;
    float bv = bias ? bias[col] : 0.0f;
#pragma unroll
    for (int i = 0; i < 8; ++i) {
        int row = tm + i + half * 8;
        float v = acc[i] + bv;
        if (act == 1) v = (v > 0.f) ? v : 0.01f * v;
        Cm[(size_t)row * N + col] = v;
    }
}

// ---------------------------------------------------------------------------
// Helpers
// ---------------------------------------------------------------------------
__device__ __forceinline__ unsigned f2ord(float f) {
    unsigned u = __float_as_uint(f);
    return (u & 0x80000000u) ? ~u : (u | 0x80000000u);
}
__device__ __forceinline__ float ord2f(unsigned u) {
    return (u & 0x80000000u) ? __uint_as_float(u & 0x7FFFFFFFu)
                             : __uint_as_float(~u);
}

__global__ void zero_f32(float* p, int n) {
    int i = blockIdx.x * blockDim.x + threadIdx.x;
    if (i < n) p[i] = 0.f;
}
__global__ void zero_u32(unsigned* p, int n) {
    int i = blockIdx.x * blockDim.x + threadIdx.x;
    if (i < n) p[i] = 0u;
}

// ---------------------------------------------------------------------------
// BatchNorm (training-mode batch stats, eps=1e-5) + leaky_relu(0.01)
// ---------------------------------------------------------------------------
__global__ void bn_stats(const float* __restrict__ h, float* mu, float* var)
{
    __shared__ float ssum[256], ssq[256];
    int ch = blockIdx.x, t = threadIdx.x;
    float s = 0.f, q = 0.f;
    for (int n = t; n < N_NODES; n += 256) {
        float v = h[(size_t)n * C + ch];
        s += v; q += v * v;
    }
    ssum[t] = s; ssq[t] = q;
    __syncthreads();
    for (int off = 128; off > 0; off >>= 1) {
        if (t < off) { ssum[t] += ssum[t + off]; ssq[t] += ssq[t + off]; }
        __syncthreads();
    }
    if (t == 0) {
        float m = ssum[0] * (1.0f / N_NODES);
        mu[ch]  = m;
        var[ch] = ssq[0] * (1.0f / N_NODES) - m * m;   // ddof=0 (jnp.var)
    }
}

__global__ void bn_apply_leaky(const float* __restrict__ h,
                               const float* __restrict__ mu,
                               const float* __restrict__ var,
                               const float* __restrict__ gamma,
                               const float* __restrict__ beta,
                               float* __restrict__ tn)
{
    int i = blockIdx.x * blockDim.x + threadIdx.x;
    if (i >= N_NODES * C) return;
    int c = i & (C - 1);
    float v = (h[i] - mu[c]) * rsqrtf(var[c] + 1e-5f) * gamma[c] + beta[c];
    tn[i] = (v > 0.f) ? v : 0.01f * v;
}

// ---------------------------------------------------------------------------
// GATv2 edge phase.  One wave per edge; xl/xr rows are L2-resident (16 MB).
// ---------------------------------------------------------------------------
__global__ void edge_scores(const float* __restrict__ xl,
                            const float* __restrict__ xr,
                            const int* __restrict__ ei,
                            const float* __restrict__ att,
                            float* __restrict__ e, unsigned* __restrict__ mord)
{
    int w = (blockIdx.x * blockDim.x + threadIdx.x) >> 5;
    int lane = threadIdx.x & 31;
    if (w >= E_TOT) return;
    int s, d;
    if (w < NUM_EDGES) { s = ei[w]; d = ei[NUM_EDGES + w]; }
    else               { s = d = w - NUM_EDGES; }              // self loop
    const v4f* xls = (const v4f*)(xl + (size_t)s * C);
    const v4f* xrd = (const v4f*)(xr + (size_t)d * C);
    const v4f* at  = (const v4f*)att;
    v4f a = xls[lane], b = xrd[lane], t = at[lane];
    float acc = 0.f;
#pragma unroll
    for (int j = 0; j < 4; ++j) {
        float v = a[j] + b[j];
        v = (v > 0.f) ? v : 0.2f * v;                          // slope 0.2
        acc += v * t[j];
    }
    for (int off = 16; off > 0; off >>= 1) acc += __shfl_xor(acc, off, 32);
    if (lane == 0) {
        e[w] = acc;
        atomicMax(&mord[d], f2ord(acc));                       // segment max
    }
}

__global__ void edge_exp(const int* __restrict__ ei,
                         float* __restrict__ e,
                         const unsigned* __restrict__ mord,
                         float* __restrict__ denom)
{
    int i = blockIdx.x * blockDim.x + threadIdx.x;
    if (i >= E_TOT) return;
    int d = (i < NUM_EDGES) ? ei[NUM_EDGES + i] : (i - NUM_EDGES);
    float ee = expf(e[i] - ord2f(mord[d]));
    e[i] = ee;
    atomicAdd(&denom[d], ee);
}

__global__ void edge_aggregate(const float* __restrict__ xl,
                               const int* __restrict__ ei,
                               const float* __restrict__ e,
                               const float* __restrict__ denom,
                               float* __restrict__ agg)
{
    int w = (blockIdx.x * blockDim.x + threadIdx.x) >> 5;
    int lane = threadIdx.x & 31;
    if (w >= E_TOT) return;
    int s, d;
    if (w < NUM_EDGES) { s = ei[w]; d = ei[NUM_EDGES + w]; }
    else               { s = d = w - NUM_EDGES; }
    float alpha = e[w] / denom[d];
    const v4f* xls = (const v4f*)(xl + (size_t)s * C);
    v4f a = xls[lane];
    float* dst = agg + (size_t)d * C + lane * 4;
#pragma unroll
    for (int j = 0; j < 4; ++j) atomicAdd(dst + j, alpha * a[j]);
}

__global__ void residual_add(float* __restrict__ h,
                             const float* __restrict__ agg,
                             const float* __restrict__ cb)
{
    int i = blockIdx.x * blockDim.x + threadIdx.x;
    if (i >= N_NODES * C) return;
    h[i] += agg[i] + cb[i & (C - 1)];
}

// ---------------------------------------------------------------------------
// Mem-pool 1
// ---------------------------------------------------------------------------
__global__ void pack_K1(const float* __restrict__ K1, float* __restrict__ Kt,
                        float* __restrict__ kn)
{
    int i = blockIdx.x * blockDim.x + threadIdx.x;
    if (i < C * NC1P) {
        int c = i / NC1P, j = i % NC1P;
        Kt[i] = (j < NC1) ? K1[(size_t)j * C + c] : 0.f;       // [C, NC1P]
    }
    if (i < NC1P) {
        float s = 0.f;
        if (i < NC1)
            for (int c = 0; c < C; ++c) { float v = K1[(size_t)i * C + c]; s += v * v; }
        kn[i] = s;
    }
}

__global__ void mempool1_S(const float* __restrict__ h,
                           const float* __restrict__ dots,
                           const float* __restrict__ kn,
                           float* __restrict__ S1)
{
    int n = blockIdx.x * blockDim.x + threadIdx.x;
    if (n >= N_NODES) return;
    const float* hr = h + (size_t)n * C;
    float xn = 0.f;
    for (int c = 0; c < C; ++c) xn += hr[c] * hr[c];
    float s[NC1];
    const float* dr = dots + (size_t)n * NC1P;
    for (int j = 0; j < NC1; ++j) {
        float d = xn + kn[j] - 2.f * dr[j];                    // squared distance
        s[j] = 1.f / (1.f + d);                                // Student-t, tau=1
    }
    float Sk[K1N];
    for (int k = 0; k < K1N; ++k) Sk[k] = 0.f;
    for (int hh = 0; hh < H1; ++hh) {
        float sum = 0.f;
        for (int k = 0; k < K1N; ++k) sum += s[hh * K1N + k];
        float inv = 1.f / sum;
        for (int k = 0; k < K1N; ++k) Sk[k] += s[hh * K1N + k] * inv;
    }
    for (int k = 0; k < K1N; ++k)
        S1[(size_t)n * K1N + k] = Sk[k] * (1.f / H1);          // mean over heads
}

__global__ void pool1(const float* __restrict__ S1, const float* __restrict__ h,
                      float* __restrict__ xp)
{
    int b = blockIdx.x, k = blockIdx.y, c = threadIdx.x;       // block 128
    int base = b * PER_G;
    float acc = 0.f;
    for (int n = 0; n < PER_G; ++n)
        acc += S1[(size_t)(base + n) * K1N + k] * h[(size_t)(base + n) * C + c];
    xp[((size_t)b * K1N + k) * C + c] = acc;
}

// ---------------------------------------------------------------------------
// Mem-pool 2: K=1 per head => S2 == 1 exactly (s/s), KL(S2) == 0 exactly.
// x2[b] = (sum_k x1[b,k,:]) @ Wm2
// ---------------------------------------------------------------------------
__global__ void pool2_gemm(const float* __restrict__ x1,
                           const float* __restrict__ Wm2,
                           float* __restrict__ x2)
{
    __shared__ float xs[C_MID];
    int b = blockIdx.x, t = threadIdx.x;                       // block 80
    float s = 0.f;
    for (int k = 0; k < K1N; ++k) s += x1[((size_t)b * K1N + k) * C_MID + t];
    xs[t] = s;
    __syncthreads();
    if (t < C_OUT) {
        float acc = 0.f;
        for (int q = 0; q < C_MID; ++q) acc += xs[q] * Wm2[(size_t)q * C_OUT + t];
        x2[(size_t)b * C_OUT + t] = acc;
    }
}

__global__ void log_softmax_out(const float* __restrict__ x2, float* __restrict__ out)
{
    __shared__ float red[C_OUT];
    int b = blockIdx.x, t = threadIdx.x;                       // block 64
    float v = x2[(size_t)b * C_OUT + t];
    red[t] = v;
    __syncthreads();
    for (int off = 32; off > 0; off >>= 1) {
        if (t < off) red[t] = fmaxf(red[t], red[t + off]);
        __syncthreads();
    }
    float m = red[0];
    __syncthreads();
    red[t] = expf(v - m);
    __syncthreads();
    for (int off = 32; off > 0; off >>= 1) {
        if (t < off) red[t] += red[t + off];
        __syncthreads();
    }
    out[(size_t)b * C_OUT + t] = v - (m + logf(red[0]));
}

// ---------------------------------------------------------------------------
// KL loss for S1
// ---------------------------------------------------------------------------
__global__ void colsum_k(const float* __restrict__ S1, float* __restrict__ cs)
{
    int i = blockIdx.x * blockDim.x + threadIdx.x;
    if (i >= NUM_GRAPHS * K1N) return;
    int b = i / K1N, k = i % K1N;
    float s = 0.f;
    for (int n = 0; n < PER_G; ++n) s += S1[(size_t)(b * PER_G + n) * K1N + k];
    cs[i] = s;
}

__global__ void kl1_k(const float* __restrict__ S1, const float* __restrict__ cs,
                      float* __restrict__ loss)
{
    int i = blockIdx.x * blockDim.x + threadIdx.x;
    if (i >= N_NODES) return;
    int b = i / PER_G;
    float s2[K1N];
    float den = 0.f, ssum = 0.f;
    for (int k = 0; k < K1N; ++k) {
        float sv = S1[(size_t)i * K1N + k];
        float v  = sv * sv / cs[b * K1N + k];
        s2[k] = v; den += v; ssum += sv;
    }
    if (ssum == 0.f) den = 1.f;
    float acc = 0.f;
    for (int k = 0; k < K1N; ++k) {
        float P  = fmaxf(s2[k] / den, KL_EPS);
        float Sc = fmaxf(S1[(size_t)i * K1N + k], KL_EPS);
        acc += P * (logf(P) - logf(Sc));
    }
    for (int off = 16; off > 0; off >>= 1) acc += __shfl_xor(acc, off, 32);
    if ((threadIdx.x & 31) == 0) atomicAdd(loss, acc);
}

__global__ void finalize_loss(const float* __restrict__ loss, float* __restrict__ out)
{
    if (threadIdx.x == 0 && blockIdx.x == 0)
        out[NUM_GRAPHS * C_OUT] = loss[0] * (1.f / NUM_GRAPHS);  // KL2 == 0 exactly
}

// ---------------------------------------------------------------------------
extern "C" void kernel_launch(void* const* d_in, const int* in_sizes, int n_in,
                              void* d_out, int out_size, void* d_ws, size_t ws_size,
                              hipStream_t stream)
{
    const float* x        = (const float*)d_in[0];
    const int*   ei       = (const int*)d_in[1];
    // d_in[2] = batch (implicit: equal-size graphs)
    const float* W_in     = (const float*)d_in[3];
    const float* b_in     = (const float*)d_in[4];
    const float* bn_gamma = (const float*)d_in[5];
    const float* bn_beta  = (const float*)d_in[6];
    const float* Wl       = (const float*)d_in[7];
    const float* bl       = (const float*)d_in[8];
    const float* Wr       = (const float*)d_in[9];
    const float* br       = (const float*)d_in[10];
    const float* att      = (const float*)d_in[11];
    const float* cb       = (const float*)d_in[12];
    const float* K1       = (const float*)d_in[13];
    const float* Wm1      = (const float*)d_in[14];
    // d_in[15] = K2 (unused: S2 == 1 exactly, KL2 == 0 exactly)
    const float* Wm2      = (const float*)d_in[16];
    float* out = (float*)d_out;

    // Workspace layout (floats); post-layer buffers alias dead tn/xl regions.
    float* ws = (float*)d_ws;
    const size_t NHC = (size_t)N_NODES * C;                    // 4,194,304
    float*    h     = ws;
    float*    tn    = ws + NHC;
    float*    xl    = ws + 2 * NHC;
    float*    xr    = ws + 3 * NHC;                            // aliased as agg
    float*    agg   = xr;
    float*    e     = ws + 4 * NHC;                            // 557,056
    unsigned* mord  = (unsigned*)(e + E_TOT);                  // 32,768
    float*    denom = (float*)(mord + N_NODES);                // 32,768
    float*    mu    = denom + N_NODES;                         // 128
    float*    var   = mu + C;                                  // 128
    // post-layer aliases:
    float* dots = tn;                                          // 32768*64
    float* S1   = tn + (size_t)N_NODES * NC1P;                 // 32768*10
    float* xp   = S1 + (size_t)N_NODES * K1N;                  // 32*10*128
    float* x1   = xp + (size_t)NUM_GRAPHS * K1N * C;           // 32*10*80
    float* x2   = x1 + (size_t)NUM_GRAPHS * K1N * C_MID;       // 32*64
    float* cs   = x2 + (size_t)NUM_GRAPHS * C_OUT;             // 320
    float* loss = cs + NUM_GRAPHS * K1N;                       // 1
    float* Kt   = xl;                                          // 128*64
    float* kn   = xl + C * NC1P;                               // 64

    const int NODE_ELEMS = N_NODES * C;
    const int GB = (NODE_ELEMS + 255) / 256;                   // elementwise grid
    const int EW = (E_TOT * 32 + 255) / 256;                   // wave-per-edge grid
    const int ET = (E_TOT + 255) / 256;                        // thread-per-edge grid

    // h = x @ W_in + b_in      (32768 x 144 x 128, fp32 WMMA, 64x32 macro-tiles)
    {
        int tiles = (N_NODES / 64) * (C / 32);                 // 2048
        wmma_gemm_f32_t64x32<<<(tiles + 7) / 8, 256, 0, stream>>>(x, W_in, b_in, h,
                                                                  N_NODES, D_IN, C);
    }

    for (int l = 0; l < N_LAYERS; ++l) {
        bn_stats<<<C, 256, 0, stream>>>(h, mu, var);
        bn_apply_leaky<<<GB, 256, 0, stream>>>(h, mu, var,
                                               bn_gamma + l * C, bn_beta + l * C, tn);
        int tiles = (N_NODES / 64) * (C / 32);                 // 2048
        wmma_gemm_f32_t64x32<<<(tiles + 7) / 8, 256, 0, stream>>>(tn, Wl + (size_t)l * C * C,
                                                                  bl + l * C, xl, N_NODES, C, C);
        wmma_gemm_f32_t64x32<<<(tiles + 7) / 8, 256, 0, stream>>>(tn, Wr + (size_t)l * C * C,
                                                                  br + l * C, xr, N_NODES, C, C);
        zero_u32<<<(N_NODES + 255) / 256, 256, 0, stream>>>(mord, N_NODES);
        zero_f32<<<(N_NODES + 255) / 256, 256, 0, stream>>>(denom, N_NODES);
        edge_scores<<<EW, 256, 0, stream>>>(xl, xr, ei, att + l * C, e, mord);
        edge_exp<<<ET, 256, 0, stream>>>(ei, e, mord, denom);
        zero_f32<<<GB, 256, 0, stream>>>(agg, NODE_ELEMS);     // xr dead from here
        edge_aggregate<<<EW, 256, 0, stream>>>(xl, ei, e, denom, agg);
        residual_add<<<GB, 256, 0, stream>>>(h, agg, cb + l * C);
    }

    // Mem-pool 1: dots = h @ K1t (32768 x 128 x 64, fp32 WMMA, 64x32 macro-tiles)
    pack_K1<<<(C * NC1P + 255) / 256, 256, 0, stream>>>(K1, Kt, kn);
    {
        int tiles = (N_NODES / 64) * (NC1P / 32);              // 1024
        wmma_gemm_f32_t64x32<<<(tiles + 7) / 8, 256, 0, stream>>>(h, Kt, nullptr, dots,
                                                                  N_NODES, C, NC1P);
    }
    mempool1_S<<<(N_NODES + 255) / 256, 256, 0, stream>>>(h, dots, kn, S1);
    pool1<<<dim3(NUM_GRAPHS, K1N), C, 0, stream>>>(S1, h, xp);
    {   // x1 = leaky_relu(xp @ Wm1)   (320 x 128 x 80, fp32 WMMA, generic tiles)
        int tiles = (NUM_GRAPHS * K1N / 16) * (C_MID / 16);    // 100
        wmma_gemm_f32<<<(tiles + 7) / 8, 256, 0, stream>>>(xp, Wm1, nullptr, x1,
                                                           NUM_GRAPHS * K1N, C, C_MID, 1);
    }
    pool2_gemm<<<NUM_GRAPHS, C_MID, 0, stream>>>(x1, Wm2, x2);
    log_softmax_out<<<NUM_GRAPHS, C_OUT, 0, stream>>>(x2, out);

    // KL(S1) (KL(S2)==0 exactly since K2 has a single cluster per head)
    colsum_k<<<2, 256, 0, stream>>>(S1, cs);
    zero_f32<<<1, 32, 0, stream>>>(loss, 1);
    kl1_k<<<(N_NODES + 255) / 256, 256, 0, stream>>>(S1, cs, loss);
    finalize_loss<<<1, 32, 0, stream>>>(loss, out);
}